// RoIAlignAvg_64974265254146
// MI455X (gfx1250) — compile-verified
//
#include <hip/hip_runtime.h>
#include <hip/hip_bf16.h>

typedef float v2f __attribute__((ext_vector_type(2)));
typedef float v8f __attribute__((ext_vector_type(8)));

#define FEAT_N 8
#define FEAT_C 256
#define FEAT_H 50
#define FEAT_W 68
#define HW (FEAT_H * FEAT_W)            // 3400
#define CHW (FEAT_C * HW)               // 870400
#define SSTRIDE 264                     // LDS row stride in floats; 264 % 64 == 8 -> conflict-free WMMA A reads
#define OUT_PER_ROI (FEAT_C * 49)       // 12544

// Pooling weight: Out[o] = 0.25 * sum over sample points p=(i,j) with i in {oh,oh+1}, j in {ow,ow+1}
__device__ __forceinline__ float pool_w(int valid_o, int oh, int ow, int p) {
    int i = p >> 3;
    int j = p & 7;
    unsigned di = (unsigned)(i - oh);
    unsigned dj = (unsigned)(j - ow);
    return (valid_o && di <= 1u && dj <= 1u) ? 0.25f : 0.0f;
}

__global__ __launch_bounds__(256)
void roialign_avg_wmma_kernel(const float* __restrict__ feat,
                              const float* __restrict__ rois,
                              float* __restrict__ out) {
    extern __shared__ float S[];        // [64 sample points][SSTRIDE] floats

    const int t = threadIdx.x;
    const int r = blockIdx.x;

    // ---------------- Phase 1: bilinear sampling into LDS ----------------
    // thread t owns sample point p = t & 63, channels c = (t>>6), +4 stride
    const float* roi = rois + (size_t)r * 5;
    const int   b  = (int)roi[0];
    const float x1 = roi[1] * 0.0625f;
    const float y1 = roi[2] * 0.0625f;
    const float x2 = roi[3] * 0.0625f;
    const float y2 = roi[4] * 0.0625f;
    const float roi_w = fmaxf(x2 - x1, 0.0f);
    const float roi_h = fmaxf(y2 - y1, 0.0f);
    const float bin_h = roi_h * (1.0f / 7.0f);
    const float bin_w = roi_w * (1.0f / 7.0f);

    const int p = t & 63;
    const int si = p >> 3;              // sample row (h direction), 0..7
    const int sj = p & 7;               // sample col (w direction), 0..7

    const float h = y1 + (float)si * bin_h;
    const float w = x1 + (float)sj * bin_w;
    const bool inb = (h >= 0.0f) && (h < (float)FEAT_H) &&
                     (w >= 0.0f) && (w < (float)FEAT_W);
    const int y0 = (int)fminf(fmaxf(floorf(h), 0.0f), (float)(FEAT_H - 2));
    const int x0 = (int)fminf(fmaxf(floorf(w), 0.0f), (float)(FEAT_W - 2));
    const float ly = h - (float)y0;
    const float lx = w - (float)x0;
    const float m   = inb ? 1.0f : 0.0f;
    const float w00 = (1.0f - ly) * (1.0f - lx) * m;
    const float w01 = (1.0f - ly) * lx * m;
    const float w10 = ly * (1.0f - lx) * m;
    const float w11 = ly * lx * m;

    const float* fb = feat + (size_t)b * CHW + (size_t)(y0 * FEAT_W + x0);
    const int c0 = t >> 6;              // 0..3
    #pragma unroll 4
    for (int c = c0; c < FEAT_C; c += 4) {
        const float* fp = fb + (size_t)c * HW;
        float v00 = fp[0];
        float v01 = fp[1];
        float v10 = fp[FEAT_W];
        float v11 = fp[FEAT_W + 1];
        S[p * SSTRIDE + c] = w00 * v00 + w01 * v01 + w10 * v10 + w11 * v11;
    }

    __syncthreads();

    // ---------------- Phase 2: pooling as WMMA ----------------
    // Out^T[c, o] = sum_p S[p][c] * P[o, p]
    // A (16x4 f32): lane%16 = M (channel row), VGPR v = K = v + 2*(lane>>4)
    // B (4x16 f32): lane%16 = N (pooled pos),  VGPR v = K = v + 2*(lane>>4)
    // D (16x16 f32): lane%16 = N, VGPR v -> M = v + 8*(lane>>4)
    const int lane = t & 31;
    const int lo   = lane & 15;
    const int hi   = lane >> 4;
    const int wave = t >> 5;            // 0..7
    const int mt0  = wave * 2;          // this wave's channel-tile pair
    const int cA0  = mt0 * 16 + lo;     // A column (channel) for tile 0
    const int cA1  = cA0 + 16;          // for tile 1

    for (int nt = 0; nt < 4; ++nt) {
        const int o  = nt * 16 + lo;    // pooled position 0..63 (valid < 49)
        const int vo = (o < 49);
        const int oh = o / 7;
        const int ow = o - oh * 7;

        v8f acc0 = {};
        v8f acc1 = {};
        #pragma unroll
        for (int k = 0; k < 16; ++k) {
            const int p0 = k * 4 + 2 * hi;
            const int p1 = p0 + 1;
            v2f Bv, A0, A1;
            Bv.x = pool_w(vo, oh, ow, p0);
            Bv.y = pool_w(vo, oh, ow, p1);
            A0.x = S[p0 * SSTRIDE + cA0];
            A0.y = S[p1 * SSTRIDE + cA0];
            A1.x = S[p0 * SSTRIDE + cA1];
            A1.y = S[p1 * SSTRIDE + cA1];
            acc0 = __builtin_amdgcn_wmma_f32_16x16x4_f32(
                false, A0, false, Bv, (short)0, acc0, false, false);
            acc1 = __builtin_amdgcn_wmma_f32_16x16x4_f32(
                false, A1, false, Bv, (short)0, acc1, false, false);
        }

        if (vo) {
            const size_t base = (size_t)r * OUT_PER_ROI + (size_t)o;
            const int cD0 = mt0 * 16 + 8 * hi;   // D-tile M = v + 8*hi
            const int cD1 = cD0 + 16;
            #pragma unroll
            for (int v = 0; v < 8; ++v) {
                out[base + (size_t)(cD0 + v) * 49] = acc0[v];
                out[base + (size_t)(cD1 + v) * 49] = acc1[v];
            }
        }
    }
}

extern "C" void kernel_launch(void* const* d_in, const int* in_sizes, int n_in,
                              void* d_out, int out_size, void* d_ws, size_t ws_size,
                              hipStream_t stream) {
    const float* features = (const float*)d_in[0];
    const float* rois     = (const float*)d_in[1];
    float* out            = (float*)d_out;

    const int R = in_sizes[1] / 5;                    // 4000 ROIs
    const size_t lds_bytes = (size_t)64 * SSTRIDE * sizeof(float);  // 67584 B

    roialign_avg_wmma_kernel<<<dim3(R), dim3(256), lds_bytes, stream>>>(
        features, rois, out);
}